// DynamicAttentionModel_60249801229148
// MI455X (gfx1250) — compile-verified
//
#include <hip/hip_runtime.h>
#include <hip/hip_bf16.h>

#define B_  16
#define D_  2048
#define NB_ 8
#define Z_  64
#define H_  256
#define C_  10
#define SPLITJ 4
#define JCHUNK (D_ / SPLITJ)   // 512

typedef __attribute__((ext_vector_type(2))) float v2f;
typedef __attribute__((ext_vector_type(8))) float v8f;

// ---------------- K1: hypernetwork -> mixing coefficients [16,8] ----------------
__global__ __launch_bounds__(256) void coeffs_kernel(
    const int* __restrict__ seeds, const float* __restrict__ seed_emb,
    const float* __restrict__ static_coeffs,
    const float* __restrict__ hw1, const float* __restrict__ hb1,
    const float* __restrict__ hw2, const float* __restrict__ hb2,
    float* __restrict__ coeffs_out)
{
    __shared__ float h[B_][H_];
    __shared__ float l8[B_][NB_];
    const int t = threadIdx.x;

    // h = relu(z @ hw1 + hb1); thread t owns hidden unit t for all 16 rows
    for (int b = 0; b < B_; ++b) {
        float acc = hb1[t];
        const float* z = seed_emb + (size_t)seeds[b] * Z_;
        #pragma unroll 8
        for (int j = 0; j < Z_; ++j) acc += z[j] * hw1[j * H_ + t];
        h[b][t] = acc > 0.f ? acc : 0.f;
    }
    __syncthreads();

    if (t < B_ * NB_) {               // logits8 = h @ hw2 + hb2
        const int b = t >> 3, nb = t & 7;
        float acc = hb2[nb];
        #pragma unroll 8
        for (int u = 0; u < H_; ++u) acc += h[b][u] * hw2[u * NB_ + nb];
        l8[b][nb] = acc;
    }
    __syncthreads();

    if (t < B_) {                     // per-row softmax + static softmax
        float m = l8[t][0];
        #pragma unroll
        for (int i = 1; i < NB_; ++i) m = fmaxf(m, l8[t][i]);
        float e[NB_], s = 0.f;
        #pragma unroll
        for (int i = 0; i < NB_; ++i) { e[i] = __expf(l8[t][i] - m); s += e[i]; }
        float sm = static_coeffs[0];
        #pragma unroll
        for (int i = 1; i < NB_; ++i) sm = fmaxf(sm, static_coeffs[i]);
        float se[NB_], ss = 0.f;
        #pragma unroll
        for (int i = 0; i < NB_; ++i) { se[i] = __expf(static_coeffs[i] - sm); ss += se[i]; }
        #pragma unroll
        for (int i = 0; i < NB_; ++i)
            coeffs_out[t * NB_ + i] = se[i] / ss + e[i] / s;
    }
}

// ---------------- K2: partial v over one j-slice ----------------
// grid = (128 col strips, SPLITJ j-slices); 8 waves per block = one basis each.
// Each block streams a disjoint slice of wv exactly once (NT loads).
__global__ __launch_bounds__(256) void vproj_kernel(
    const float* __restrict__ features, const float* __restrict__ wv,
    const float* __restrict__ coeffs, float* __restrict__ vpart)
{
    __shared__ float ldsF[B_][JCHUNK + 3];   // row stride 515 -> conflict-free A reads
    __shared__ float ldsP[NB_][256];

    const int t    = threadIdx.x;
    const int wave = t >> 5;          // basis index i
    const int lane = t & 31;
    const int n    = lane & 15;       // A row M / B col N within tile
    const int hi   = lane >> 4;       // half-wave select
    const int kk   = hi << 1;         // K offset 0 or 2 (16x16x4 layout)
    const int k0   = blockIdx.x << 4; // output column strip
    const int jc   = blockIdx.y * JCHUNK;

    const float cf = coeffs[n * NB_ + wave];   // folded into A operand
    v8f acc = {};

    // stage features[:, jc:jc+512] -> LDS (coalesced float4), once per block
    for (int e = t * 4; e < B_ * JCHUNK; e += 256 * 4) {
        const int r = e / JCHUNK;
        const int c = e - r * JCHUNK;
        const float4 f4 = *(const float4*)(features + (size_t)r * D_ + jc + c);
        ldsF[r][c + 0] = f4.x; ldsF[r][c + 1] = f4.y;
        ldsF[r][c + 2] = f4.z; ldsF[r][c + 3] = f4.w;
    }
    __syncthreads();

    const float* wptr = wv + ((size_t)wave * D_ + (size_t)(jc + kk)) * D_ + k0 + n;
    #pragma unroll 8
    for (int jj = 0; jj < JCHUNK; jj += 4) {
        v2f a, bm;
        a.x  = cf * ldsF[n][jj + kk];              // A: M=n, K=kk
        a.y  = cf * ldsF[n][jj + kk + 1];          // A: M=n, K=kk+1
        bm.x = __builtin_nontemporal_load(wptr);       // B: K=kk,   N=n
        bm.y = __builtin_nontemporal_load(wptr + D_);  // B: K=kk+1, N=n
        wptr += (size_t)4 * D_;
        acc = __builtin_amdgcn_wmma_f32_16x16x4_f32(
            false, a, false, bm, (short)0, acc, false, false);
    }

    // scatter D tile (8 VGPRs, C/D layout) to logical [row][col], reduce over bases
    #pragma unroll
    for (int v = 0; v < 8; ++v) {
        const int row = v + (hi << 3);
        ldsP[wave][row * 16 + n] = acc[v];
    }
    __syncthreads();
    {
        const int row = t >> 4, col = t & 15;
        float s = 0.f;
        #pragma unroll
        for (int i = 0; i < NB_; ++i) s += ldsP[i][t];
        vpart[((size_t)blockIdx.y * B_ + row) * D_ + k0 + col] = s;
    }
}

// ---------------- K3: logits = (sum of j-slice partials of v) @ cw + cb ; attn = ones ----
__global__ __launch_bounds__(192) void head_kernel(
    const float* __restrict__ vpart, const float* __restrict__ cw,
    const float* __restrict__ cb, float* __restrict__ out)
{
    const int t = threadIdx.x;
    if (t < B_ * C_) {
        const int b = t / C_, c = t - b * C_;
        float acc = cb[c];
        const float* vp = vpart + (size_t)b * D_;
        #pragma unroll 4
        for (int k = 0; k < D_; ++k) {
            float vsum = 0.f;
            #pragma unroll
            for (int p = 0; p < SPLITJ; ++p)
                vsum += vp[(size_t)p * B_ * D_ + k];
            acc += vsum * cw[k * C_ + c];
        }
        out[t] = acc;
    } else if (t < B_ * C_ + B_) {
        out[t] = 1.0f;   // softmax over a singleton axis is exactly 1
    }
}

extern "C" void kernel_launch(void* const* d_in, const int* in_sizes, int n_in,
                              void* d_out, int out_size, void* d_ws, size_t ws_size,
                              hipStream_t stream) {
    const float* features       = (const float*)d_in[0];
    const int*   seeds          = (const int*)  d_in[1];
    const float* seed_emb       = (const float*)d_in[2];
    const float* static_coeffs  = (const float*)d_in[3];
    const float* hw1            = (const float*)d_in[4];
    const float* hb1            = (const float*)d_in[5];
    const float* hw2            = (const float*)d_in[6];
    const float* hb2            = (const float*)d_in[7];
    // d_in[8]=wq, d_in[9]=wk are dead: seq_len==1 softmax makes attention identity.
    const float* wv             = (const float*)d_in[10];
    const float* cw             = (const float*)d_in[11];
    const float* cb             = (const float*)d_in[12];

    float* coeffs_ws = (float*)d_ws;         // 128 floats used, padded to 256
    float* v_ws      = coeffs_ws + 256;      // SPLITJ * 16 * 2048 floats (512 KB)

    hipLaunchKernelGGL(coeffs_kernel, dim3(1), dim3(256), 0, stream,
                       seeds, seed_emb, static_coeffs, hw1, hb1, hw2, hb2, coeffs_ws);
    hipLaunchKernelGGL(vproj_kernel, dim3(D_ / 16, SPLITJ), dim3(256), 0, stream,
                       features, wv, coeffs_ws, v_ws);
    hipLaunchKernelGGL(head_kernel, dim3(1), dim3(192), 0, stream,
                       v_ws, cw, cb, (float*)d_out);
}